// SkimageHOG_29901562315054
// MI455X (gfx1250) — compile-verified
//
#include <hip/hip_runtime.h>
#include <hip/hip_bf16.h>

#define IMG   512
#define TILE_W 64
#define TILE_H 32
#define LDS_W  72   // 64 + 4 halo each side (16B-aligned segments)
#define LDS_H  34   // 32 + 1 halo each side
#define NCELL  64   // cells per image row/col
#define NBLK   63   // blocks per image row/col
#define ORI    9

// ---------------------------------------------------------------------------
// Kernel 1: per-cell orientation histograms.
// Grid: (8 tile-cols, 16 tile-rows, 64 images), 256 threads (8 wave32).
// Stages a 34x72 fp32 tile into LDS via CDNA5 async global->LDS B128 copies,
// then each thread reduces an 8-pixel cell column; 8 lanes/cell reduced with
// __shfl_xor; lane0 of each cell writes 9 floats (already divided by 64).
// ---------------------------------------------------------------------------
__global__ __launch_bounds__(256)
void hog_cell_hist(const float* __restrict__ img, float* __restrict__ hist)
{
    __shared__ __align__(16) float tile[LDS_H * LDS_W];

    const int tid     = threadIdx.x;
    const int tile_c0 = blockIdx.x * TILE_W;
    const int tile_r0 = blockIdx.y * TILE_H;
    const int b       = blockIdx.z;
    const float* base = img + (size_t)b * (IMG * IMG);

    // ---- async stage: 34 rows x 18 B128 segments = 612 transfers ----
    const unsigned lds_base = (unsigned)(uintptr_t)&tile[0]; // low 32b of generic ptr == LDS offset
    for (int i = tid; i < LDS_H * (LDS_W / 4); i += 256) {
        int row = i / (LDS_W / 4);
        int seg = i - row * (LDS_W / 4);
        int gr  = tile_r0 - 1 + row;
        gr = min(max(gr, 0), IMG - 1);            // clamped halo only feeds border-zeroed grads
        int gc  = tile_c0 - 4 + seg * 4;
        gc = min(max(gc, 0), IMG - 4);            // keeps 16B alignment (multiples of 4 floats)
        unsigned long long src = (unsigned long long)(const void*)(base + (size_t)gr * IMG + gc);
        unsigned lds_addr = lds_base + (unsigned)(row * LDS_W + seg * 4) * 4u;
        asm volatile("global_load_async_to_lds_b128 %0, %1, off"
                     :: "v"(lds_addr), "v"(src) : "memory");
    }
    asm volatile("s_wait_asynccnt 0" ::: "memory"); // wait this wave's async copies
    __syncthreads();                                // then all waves' copies are visible

    // ---- per-thread: one 8-pixel column of one cell ----
    const int pcol = tid & 7;          // lane bits [2:0]
    const int ccol = (tid >> 3) & 7;
    const int crow = tid >> 6;         // 0..3
    const int lc   = ccol * 8 + pcol + 4;
    const int gc   = tile_c0 + ccol * 8 + pcol;
    const bool col_border = (gc == 0) || (gc == IMG - 1);

    float h[ORI];
#pragma unroll
    for (int k = 0; k < ORI; ++k) h[k] = 0.0f;

#pragma unroll
    for (int pr = 0; pr < 8; ++pr) {
        const int lrow = crow * 8 + pr + 1;
        const int grow = tile_r0 + crow * 8 + pr;
        float y = (grow == 0 || grow == IMG - 1) ? 0.0f
                : tile[(lrow + 1) * LDS_W + lc] - tile[(lrow - 1) * LDS_W + lc];
        float x = col_border ? 0.0f
                : tile[lrow * LDS_W + lc + 1] - tile[lrow * LDS_W + lc - 1];
        if (y < 0.0f) { y = -y; x = -x; }          // fold angle into [0,180)
        const float mag = sqrtf(fmaf(x, x, y * y));
        // bin = #{k : angle >= 20k deg}  via sign of y*cos(th) - x*sin(th)
        int bin = 0;
        bin += (y *  0.93969262f >= x * 0.34202014f);
        bin += (y *  0.76604444f >= x * 0.64278761f);
        bin += (y *  0.50000000f >= x * 0.86602540f);
        bin += (y *  0.17364818f >= x * 0.98480775f);
        bin += (y * -0.17364818f >= x * 0.98480775f);
        bin += (y * -0.50000000f >= x * 0.86602540f);
        bin += (y * -0.76604444f >= x * 0.64278761f);
        bin += (y * -0.93969262f >= x * 0.34202014f);
        if (y == 0.0f && x < 0.0f) bin = 0;        // atan2(0,-)=180deg -> mod 180 -> bin 0
#pragma unroll
        for (int k = 0; k < ORI; ++k) h[k] += (bin == k) ? mag : 0.0f;
    }

    // ---- reduce the 8 column-lanes of each cell (lanes differ in bits 0..2) ----
#pragma unroll
    for (int k = 0; k < ORI; ++k) {
        h[k] += __shfl_xor(h[k], 1, 32);
        h[k] += __shfl_xor(h[k], 2, 32);
        h[k] += __shfl_xor(h[k], 4, 32);
    }
    if (pcol == 0) {
        const int cr = blockIdx.y * 4 + crow;
        const int cc = blockIdx.x * 8 + ccol;
        float* dst = hist + (((size_t)b * NCELL + cr) * NCELL + cc) * ORI;
#pragma unroll
        for (int k = 0; k < ORI; ++k) dst[k] = h[k] * (1.0f / 64.0f);
    }
}

// ---------------------------------------------------------------------------
// Kernel 2: 2x2-cell sliding blocks with L2-Hys normalization.
// One thread per block; output laid out exactly as the reference ravel:
// ((b*63+br)*63+bc)*36 + (i*2+j)*9 + k.
// ---------------------------------------------------------------------------
__global__ __launch_bounds__(256)
void hog_block_norm(const float* __restrict__ hist, float* __restrict__ out)
{
    const int NB = NBLK * NBLK;
    int t = blockIdx.x * 256 + threadIdx.x;
    if (t >= 64 * NB) return;
    const int b  = t / NB;
    const int r  = t - b * NB;
    const int br = r / NBLK;
    const int bc = r - br * NBLK;

    const float* h0 = hist + (((size_t)b * NCELL + br) * NCELL + bc) * ORI;
    float v[4 * ORI];
    float ss = 0.0f;
#pragma unroll
    for (int i = 0; i < 2; ++i)
#pragma unroll
        for (int j = 0; j < 2; ++j)
#pragma unroll
            for (int k = 0; k < ORI; ++k) {
                float x = h0[((size_t)i * NCELL + j) * ORI + k];
                v[(i * 2 + j) * ORI + k] = x;
                ss = fmaf(x, x, ss);
            }
    const float inv1 = 1.0f / sqrtf(ss + 1e-10f);   // eps^2, eps = 1e-5
    float ss2 = 0.0f;
#pragma unroll
    for (int m = 0; m < 4 * ORI; ++m) {
        float x = fminf(v[m] * inv1, 0.2f);
        v[m] = x;
        ss2 = fmaf(x, x, ss2);
    }
    const float inv2 = 1.0f / sqrtf(ss2 + 1e-10f);
    float* dst = out + (size_t)t * (4 * ORI);
#pragma unroll
    for (int m = 0; m < 4 * ORI; ++m) dst[m] = v[m] * inv2;
}

extern "C" void kernel_launch(void* const* d_in, const int* in_sizes, int n_in,
                              void* d_out, int out_size, void* d_ws, size_t ws_size,
                              hipStream_t stream)
{
    const float* x   = (const float*)d_in[0];   // (64,1,512,512) fp32
    float* out       = (float*)d_out;           // 64*63*63*36 fp32
    float* hist      = (float*)d_ws;            // 64*64*64*9 fp32 = 9.4 MB scratch

    dim3 grid1(IMG / TILE_W, IMG / TILE_H, 64); // (8,16,64)
    hog_cell_hist<<<grid1, 256, 0, stream>>>(x, hist);

    const int total = 64 * NBLK * NBLK;
    hog_block_norm<<<(total + 255) / 256, 256, 0, stream>>>(hist, out);
}